// OuterEquivariantMap_16097537425914
// MI455X (gfx1250) — compile-verified
//
#include <hip/hip_runtime.h>

// ---------------------------------------------------------------------------
// MinkowskiAvgPool(16) -> Conv3d(3->64,k=3,SAME) -> PoolingTranspose(16)
//
// Roofline: output is 2M x 64 fp32 = 512MB streamed writes -> ~24us at
// 23.3 TB/s HBM; that stream is marked non-temporal (TH=NT) so it does not
// evict the hot 8MB conv grid from the 192MB L2 that the gather re-reads
// 32M times. The conv GEMM is [32768 x 81] x [81 x 64] ~ 340 MFLOP,
// negligible; it runs on V_WMMA_F32_16X16X4_F32 (fp32 WMMA) so precision
// matches the fp32 reference. Pool stats (512KB) and the 21KB weight matrix
// live in L2 / LDS.
// ---------------------------------------------------------------------------

typedef __attribute__((ext_vector_type(2))) float v2f;
typedef __attribute__((ext_vector_type(4))) float v4f;
typedef __attribute__((ext_vector_type(8))) float v8f;

#define G      32
#define NVOX   (G * G * G)          // 32768
#define CIN    3
#define COUT   64
#define KTOT   (27 * CIN)           // 81
#define KPAD   84                   // pad K to multiple of 4 for 16x16x4 steps

// workspace layout (float offsets)
#define WS_SUMS 0                       // [NVOX*3]
#define WS_CNT  (NVOX * CIN)            // [NVOX]
#define WS_AVG  (WS_CNT + NVOX)         // [NVOX*3]
#define WS_BMAT (WS_AVG + NVOX * CIN)   // [KPAD*COUT]
#define WS_CONV (WS_BMAT + KPAD * COUT) // [NVOX*COUT]

// ---------------- 1) zero accumulators -------------------------------------
__global__ void mk_zero_kernel(float* __restrict__ p, int n) {
    int i = blockIdx.x * blockDim.x + threadIdx.x;
    if (i < n) p[i] = 0.0f;
}

// ---------------- 2) scatter points into coarse voxels ----------------------
__global__ void mk_scatter_kernel(const int* __restrict__ coords,
                                  const float* __restrict__ feats,
                                  float* __restrict__ sums,
                                  float* __restrict__ cnt, int n) {
    int p = blockIdx.x * blockDim.x + threadIdx.x;
    if (p >= n) return;
    int x = coords[3 * p + 0] >> 4;      // STRIDE = 16
    int y = coords[3 * p + 1] >> 4;
    int z = coords[3 * p + 2] >> 4;
    int idx = (x * G + y) * G + z;
    // fp32 hardware atomics (global_atomic_add_f32); order-independent sums.
    unsafeAtomicAdd(&sums[3 * idx + 0], feats[3 * p + 0]);
    unsafeAtomicAdd(&sums[3 * idx + 1], feats[3 * p + 1]);
    unsafeAtomicAdd(&sums[3 * idx + 2], feats[3 * p + 2]);
    unsafeAtomicAdd(&cnt[idx], 1.0f);
}

// ---------------- 3) averages ----------------------------------------------
__global__ void mk_avg_kernel(const float* __restrict__ sums,
                              const float* __restrict__ cnt,
                              float* __restrict__ avg) {
    int v = blockIdx.x * blockDim.x + threadIdx.x;
    if (v >= NVOX) return;
    float inv = 1.0f / fmaxf(cnt[v], 1.0f);
    avg[3 * v + 0] = sums[3 * v + 0] * inv;
    avg[3 * v + 1] = sums[3 * v + 1] * inv;
    avg[3 * v + 2] = sums[3 * v + 2] * inv;
}

// ---------------- 4) W [64,3,3,3,3] -> Bmat [KPAD=84][64] -------------------
// K index k = ((dx*3+dy)*3+dz)*3 + c ; rows 81..83 are zero padding.
__global__ void mk_bmat_kernel(const float* __restrict__ W,
                               float* __restrict__ bmat) {
    int i = blockIdx.x * blockDim.x + threadIdx.x;
    if (i >= KPAD * COUT) return;
    int k = i >> 6, j = i & 63;
    float v = 0.0f;
    if (k < KTOT) {
        unsigned n = (unsigned)k / 3u, c = (unsigned)k - 3u * n;
        unsigned dx = n / 9u, r = n - 9u * dx;
        unsigned dy = r / 3u, dz = r - 3u * dy;
        v = W[(((j * 3 + (int)c) * 3 + (int)dx) * 3 + (int)dy) * 3 + (int)dz];
    }
    bmat[i] = v;
}

// ---------------- 5) conv as WMMA GEMM --------------------------------------
// Branchless im2col element A[row=m][k]: zero outside grid or k>=81.
__device__ __forceinline__ float mk_loadA(const float* __restrict__ avg,
                                          int vx, int vy, int vz, int k) {
    unsigned n = (unsigned)k / 3u, c = (unsigned)k - 3u * n;
    unsigned dx = n / 9u, r = n - 9u * dx;
    unsigned dy = r / 3u, dz = r - 3u * dy;
    int xx = vx + (int)dx - 1, yy = vy + (int)dy - 1, zz = vz + (int)dz - 1;
    int valid = (k < KTOT) & ((unsigned)xx < (unsigned)G) &
                ((unsigned)yy < (unsigned)G) & ((unsigned)zz < (unsigned)G);
    int a = ((xx * G + yy) * G + zz) * 3 + (int)c;
    a = valid ? a : 0;                 // safe address, select after load
    float f = avg[a];
    return valid ? f : 0.0f;
}

__global__ __launch_bounds__(32) void mk_conv_wmma_kernel(
    const float* __restrict__ avg, const float* __restrict__ cnt,
    const float* __restrict__ bmat, float* __restrict__ convo) {
    __shared__ float sB[KPAD * COUT];  // 21 KB of the 320KB WGP LDS
    const int lane = threadIdx.x;      // wave32: block == one wave, EXEC all-1s
    for (int i = lane; i < KPAD * COUT; i += 32) sB[i] = bmat[i];
    __syncthreads();

    const int mbase = blockIdx.x * 16;          // 16-voxel M tile
    const int mrow  = mbase + (lane & 15);      // A: lanes 0-15 / 16-31 = M 0..15
    const int vx = mrow >> 10, vy = (mrow >> 5) & 31, vz = mrow & 31;
    const int khalf = (lane >> 4) * 2;          // K pair {0,1} vs {2,3} per half
    const int col = lane & 15;

    v8f acc0 = {}; v8f acc1 = {}; v8f acc2 = {}; v8f acc3 = {};

    for (int kb = 0; kb < KPAD; kb += 4) {
        const int k0 = kb + khalf;
        v2f a;                                   // ISA 32-bit A 16x4 layout
        a.x = mk_loadA(avg, vx, vy, vz, k0);
        a.y = mk_loadA(avg, vx, vy, vz, k0 + 1);
        const float* b0 = &sB[k0 * COUT + col];  // B 4x16: row K, N striped
        const float* b1 = &sB[(k0 + 1) * COUT + col];
        v2f bA; bA.x = b0[0];  bA.y = b1[0];
        v2f bB; bB.x = b0[16]; bB.y = b1[16];
        v2f bC; bC.x = b0[32]; bC.y = b1[32];
        v2f bD; bD.x = b0[48]; bD.y = b1[48];
        acc0 = __builtin_amdgcn_wmma_f32_16x16x4_f32(false, a, false, bA, (short)0, acc0, false, false);
        acc1 = __builtin_amdgcn_wmma_f32_16x16x4_f32(false, a, false, bB, (short)0, acc1, false, false);
        acc2 = __builtin_amdgcn_wmma_f32_16x16x4_f32(false, a, false, bC, (short)0, acc2, false, false);
        acc3 = __builtin_amdgcn_wmma_f32_16x16x4_f32(false, a, false, bD, (short)0, acc3, false, false);
    }

    // C/D layout: VGPR r holds M = r (lanes 0-15) or r+8 (lanes 16-31), N = lane%16.
    const int rowoff = (lane >> 4) * 8;
    float occ[8];
#pragma unroll
    for (int r = 0; r < 8; ++r)
        occ[r] = (cnt[mbase + rowoff + r] > 0.0f) ? 1.0f : 0.0f;
#pragma unroll
    for (int r = 0; r < 8; ++r) {
        float* o = convo + (size_t)(mbase + rowoff + r) * COUT + col;
        o[0]  = acc0[r] * occ[r];
        o[16] = acc1[r] * occ[r];
        o[32] = acc2[r] * occ[r];
        o[48] = acc3[r] * occ[r];
    }
}

// ---------------- 6) broadcast back to points (the 512MB stream) ------------
// thread = (point, 16B-chunk); rows are 256B aligned -> b128 load/store.
// Source rows (8MB total) stay L2-hot with default RT loads; the write-once
// 512MB output stream uses TH=NT stores so it does not thrash L2.
__global__ void mk_gather_kernel(const int* __restrict__ coords,
                                 const float* __restrict__ convo,
                                 float* __restrict__ out, int n) {
    int t = blockIdx.x * blockDim.x + threadIdx.x;
    int p = t >> 4, j4 = t & 15;
    if (p >= n) return;
    int x = coords[3 * p + 0] >> 4;
    int y = coords[3 * p + 1] >> 4;
    int z = coords[3 * p + 2] >> 4;
    int idx = (x * G + y) * G + z;
    const v4f* src = (const v4f*)(convo + (size_t)idx * COUT);
    v4f* dst = (v4f*)(out + (size_t)p * COUT);
    v4f v = src[j4];
    __builtin_nontemporal_store(v, dst + j4);   // global_store_b128 th:NT
}

// ---------------------------------------------------------------------------
extern "C" void kernel_launch(void* const* d_in, const int* in_sizes, int n_in,
                              void* d_out, int out_size, void* d_ws, size_t ws_size,
                              hipStream_t stream) {
    const int*   coords = (const int*)d_in[0];
    const float* feats  = (const float*)d_in[1];
    const float* W      = (const float*)d_in[2];
    float* out = (float*)d_out;
    float* ws  = (float*)d_ws;
    const int N = in_sizes[0] / 3;

    float* sums = ws + WS_SUMS;
    float* cnt  = ws + WS_CNT;
    float* avg  = ws + WS_AVG;
    float* bmat = ws + WS_BMAT;
    float* conv = ws + WS_CONV;

    const int nz = NVOX * CIN + NVOX;  // sums + cnt
    mk_zero_kernel<<<(nz + 255) / 256, 256, 0, stream>>>(sums, nz);
    mk_scatter_kernel<<<(N + 255) / 256, 256, 0, stream>>>(coords, feats, sums, cnt, N);
    mk_avg_kernel<<<(NVOX + 255) / 256, 256, 0, stream>>>(sums, cnt, avg);
    mk_bmat_kernel<<<(KPAD * COUT + 255) / 256, 256, 0, stream>>>(W, bmat);
    mk_conv_wmma_kernel<<<NVOX / 16, 32, 0, stream>>>(avg, cnt, bmat, conv);
    long long gt = (long long)N * 16;
    mk_gather_kernel<<<(int)((gt + 255) / 256), 256, 0, stream>>>(coords, conv, out, N);
}